// EdgeExistanceModule_84670985273481
// MI455X (gfx1250) — compile-verified
//
#include <hip/hip_runtime.h>
#include <hip/hip_bf16.h>

typedef __attribute__((ext_vector_type(2))) float v2f;
typedef __attribute__((ext_vector_type(4))) float f4;
typedef __attribute__((ext_vector_type(8))) float v8f;

// Problem sizes fixed by the reference setup.
constexpr int B_ = 4;
constexpr int N_ = 768;
constexpr int D_ = 16;
constexpr int E_ = 12288;
constexpr int NN_ = N_ * N_;          // 589824 pairs per batch
constexpr int NG_ = N_ / 16;          // 48 node-groups of 16

// ---------------------------------------------------------------------------
// Kernel 1: per-node scores s1[i]=dot(emb_i, W[:16]), s2[i]=dot(emb_i, W[16:])
// via V_WMMA_F32_16X16X4_F32: S[16x2] = Emb[16x16] x Wmat[16x2] (cols>=2 zero),
// chained as four K=4 WMMAs. One wave32 per 16 nodes.
//
// A layout (32-bit A 16x4, ISA 7.12.2): lanes 0-15 -> K={0,1},
// lanes 16-31 -> K={2,3}, M = lane&15. B mirrors with N = lane&15.
// D layout (32-bit C/D 16x16): VGPR r, lanes 0-15 -> (M=r, N=lane),
// lanes 16-31 -> (M=8+r, N=lane-16).
//
// All operand construction is branchless (v_cndmask selects) so EXEC stays
// all-1s through every WMMA, as the ISA requires.
// ---------------------------------------------------------------------------
__global__ void node_scores_wmma(const float* __restrict__ emb,
                                 const float* __restrict__ W,
                                 float* __restrict__ s1,
                                 float* __restrict__ s2) {
  const int blk  = blockIdx.x;        // 0 .. B_*NG_-1
  const int b    = blk / NG_;
  const int g    = blk % NG_;
  const int lane = threadIdx.x;       // 0..31
  const bool hi  = (lane & 16) != 0;  // false: K pair {0,1}; true: K pair {2,3}
  const int m    = lane & 15;         // row (A) / column (B) for this lane

  const f4* er4 =
      reinterpret_cast<const f4*>(emb + ((size_t)b * N_ + (size_t)g * 16 + m) * D_);
  const f4* wv = reinterpret_cast<const f4*>(W);  // wv[0..3]=W[:16], wv[4..7]=W[16:]

  const bool is0 = (m == 0);
  const bool is1 = (m == 1);

  v8f acc = {};
#pragma unroll
  for (int c = 0; c < 4; ++c) {
    const f4 e  = er4[c];        // emb[node][4c .. 4c+3]  (b128 load)
    const f4 wl = wv[c];         // W[4c .. 4c+3]          (uniform)
    const f4 wh = wv[4 + c];     // W[16+4c .. 16+4c+3]    (uniform)

    v2f a;
    a.x = hi ? e.z : e.x;        // K = 4c + 2*half
    a.y = hi ? e.w : e.y;        // K = 4c + 2*half + 1

    const float c0x = hi ? wl.z : wl.x;   // Wmat[K][0]
    const float c0y = hi ? wl.w : wl.y;
    const float c1x = hi ? wh.z : wh.x;   // Wmat[K][1]
    const float c1y = hi ? wh.w : wh.y;

    v2f bb;
    bb.x = is0 ? c0x : (is1 ? c1x : 0.0f);
    bb.y = is0 ? c0y : (is1 ? c1y : 0.0f);

    acc = __builtin_amdgcn_wmma_f32_16x16x4_f32(
        /*neg_a=*/false, a, /*neg_b=*/false, bb,
        /*c_mod=*/(short)0, acc, /*reuse_a=*/false, /*reuse_b=*/false);
  }

  // lanes with N==0 hold s1 rows, N==1 hold s2 rows (8 rows each per half)
  if (m <= 1) {
    float* dst = is0 ? s1 : s2;
    const int base = b * N_ + g * 16 + (hi ? 8 : 0);
#pragma unroll
    for (int r = 0; r < 8; ++r) dst[base + r] = acc[r];
  }
}

// ---------------------------------------------------------------------------
// Kernel 2: edge_embeddings[b, i*N+j, 0:32] = {emb[b,i], emb[b,j]}.
// 8 lanes cooperate per pair; each lane stores one aligned float4 so
// consecutive lanes hit consecutive 16B -> coalesced 128-bit NT stores
// (write-once 302MB stream; keep it out of L2).
// ---------------------------------------------------------------------------
__global__ void write_pairs(const float* __restrict__ emb,
                            float* __restrict__ out) {
  const size_t g = (size_t)blockIdx.x * blockDim.x + threadIdx.x;
  const size_t pair = g >> 3;
  const int sub = (int)(g & 7);
  if (pair >= (size_t)B_ * NN_) return;
  const int b   = (int)(pair / NN_);
  const int rem = (int)(pair % NN_);
  const int i   = rem / N_;
  const int j   = rem % N_;
  const int node = (sub < 4) ? i : j;
  const f4* src =
      reinterpret_cast<const f4*>(emb + ((size_t)b * N_ + node) * D_) +
      (sub & 3);
  f4 v = *src;
  __builtin_nontemporal_store(
      v, reinterpret_cast<f4*>(out + pair * (2 * D_)) + sub);
}

// ---------------------------------------------------------------------------
// Kernel 3: logits + zero truth. One 256-thread block per 16x16 (i,j) tile.
// Node tiles staged in LDS (rows padded to 17 floats: 17 is coprime with the
// 64 banks -> conflict-free column reads).
// ---------------------------------------------------------------------------
__global__ void logits_truth(const float* __restrict__ emb,
                             const float* __restrict__ s1ws,
                             const float* __restrict__ s2ws,
                             const float* __restrict__ bias,
                             float* __restrict__ logits,
                             float* __restrict__ truth) {
  __shared__ float ei[16][17];
  __shared__ float ej[16][17];
  __shared__ float t1[16];
  __shared__ float t2[16];

  const int blk = blockIdx.x;               // 0 .. B_*NG_*NG_-1
  const int b   = blk / (NG_ * NG_);
  const int r   = blk % (NG_ * NG_);
  const int it  = r / NG_;
  const int jt  = r % NG_;
  const int t   = threadIdx.x;              // 0..255
  const int row = t >> 4;                   // i within tile
  const int col = t & 15;                   // j within tile

  ei[row][col] = emb[((size_t)b * N_ + (size_t)it * 16 + row) * D_ + col];
  ej[row][col] = emb[((size_t)b * N_ + (size_t)jt * 16 + row) * D_ + col];
  if (t < 16)      t1[t]      = s1ws[b * N_ + it * 16 + t];
  else if (t < 32) t2[t - 16] = s2ws[b * N_ + jt * 16 + (t - 16)];
  __syncthreads();

  float diff = 0.0f;
#pragma unroll
  for (int k = 0; k < 16; ++k) diff += fabsf(ei[row][k] - ej[col][k]);

  float lg = t1[row] + t2[col] + bias[0];
  if (diff == 0.0f) lg = -10.0f;

  const size_t p = (size_t)b * NN_ + (size_t)(it * 16 + row) * N_ +
                   (size_t)jt * 16 + col;
  logits[p] = lg;
  truth[p]  = 0.0f;
}

// ---------------------------------------------------------------------------
// Kernel 4: scatter ground-truth edges. edges layout [B, 2, E] (int32).
// Duplicate indices all write the same 1.0f -> race-free.
// ---------------------------------------------------------------------------
__global__ void scatter_truth(const int* __restrict__ edges,
                              float* __restrict__ truth) {
  const int idx = blockIdx.x * blockDim.x + threadIdx.x;
  if (idx >= B_ * E_) return;
  const int b = idx / E_;
  const int e = idx % E_;
  const int i = edges[(size_t)b * 2 * E_ + e];
  const int j = edges[(size_t)b * 2 * E_ + E_ + e];
  truth[(size_t)b * NN_ + (size_t)i * N_ + j] = 1.0f;
}

extern "C" void kernel_launch(void* const* d_in, const int* in_sizes, int n_in,
                              void* d_out, int out_size, void* d_ws,
                              size_t ws_size, hipStream_t stream) {
  const float* emb   = (const float*)d_in[0];   // [B, N, D]
  const int*   edges = (const int*)d_in[1];     // [B, 2, E]
  const float* W     = (const float*)d_in[2];   // [2D]
  const float* bias  = (const float*)d_in[3];   // [1]

  float* out_edge   = (float*)d_out;                      // B*NN*2D
  float* out_logits = out_edge + (size_t)B_ * NN_ * 2 * D_;
  float* out_truth  = out_logits + (size_t)B_ * NN_;

  float* s1 = (float*)d_ws;            // B*N floats
  float* s2 = s1 + (size_t)B_ * N_;    // B*N floats

  // 1) per-node WMMA scores (tiny: 192 waves)
  node_scores_wmma<<<B_ * NG_, 32, 0, stream>>>(emb, W, s1, s2);

  // 2) logits + zero truth (depends on s1/s2 via stream order)
  logits_truth<<<B_ * NG_ * NG_, 256, 0, stream>>>(emb, s1, s2, bias,
                                                   out_logits, out_truth);

  // 3) big broadcast-store stream (independent)
  {
    const size_t total = (size_t)B_ * NN_ * 8;           // 8 lanes per pair
    const int blocks = (int)((total + 255) / 256);
    write_pairs<<<blocks, 256, 0, stream>>>(emb, out_edge);
  }

  // 4) scatter edges into truth (after zeroing, stream-ordered)
  scatter_truth<<<(B_ * E_ + 255) / 256, 256, 0, stream>>>(edges, out_truth);
}